// spatial_adaptive_operator_33724083208709
// MI455X (gfx1250) — compile-verified
//
#include <hip/hip_runtime.h>
#include <hip/hip_bf16.h>

// CDNA5 (gfx1250) wave32 WMMA types
typedef float v2f __attribute__((ext_vector_type(2)));
typedef float v8f __attribute__((ext_vector_type(8)));

// ---------------------------------------------------------------------------
// Kernel 1: off = x @ Wo^T + bo, then loc = ref + off/[W,H], written straight
// into the sampling_locations region of d_out. Uses V_WMMA_F32_16X16X4_F32.
// One wave handles a 16-row tile of the flattened (R x C) x-matrix; the 8
// real output columns (2P) are padded to 16 by duplicating Wo rows (o&7) --
// D columns are independent, the duplicates are simply never read.
// ---------------------------------------------------------------------------
__global__ __launch_bounds__(256)
void sao_off_wmma(const float* __restrict__ X,
                  const float* __restrict__ Wo,
                  const float* __restrict__ bo,
                  const int* __restrict__ Hp,
                  const int* __restrict__ Wp,
                  float* __restrict__ samploc,   // R*P*2 floats
                  int R, int C, int P, int ntiles)
{
    const int lane = threadIdx.x & 31;
    const int wid  = threadIdx.x >> 5;
    const int tile = blockIdx.x * 8 + wid;
    if (tile >= ntiles) return;                 // uniform per-wave exit

    const int Himg = *Hp, Wimg = *Wp;
    const int N = Himg * Wimg;

    const int r0    = tile * 16;
    const int mrow  = lane & 15;
    const int khalf = (lane >> 4) << 1;         // lanes 0-15 -> K+0,K+1 ; 16-31 -> K+2,K+3
    int arow = r0 + mrow; if (arow >= R) arow = R - 1;
    const int oc = mrow & 7;                    // Wo row, duplicated for padded cols 8..15

    const float* __restrict__ arowp = X  + (size_t)arow * C;
    const float* __restrict__ browp = Wo + (size_t)oc   * C;

    v8f acc = {};
    for (int kb = 0; kb < C; kb += 4) {
        const int k = kb + khalf;
        v2f a, b;
        a.x = arowp[k];
        a.y = arowp[k + 1];
        b.x = browp[k];
        b.y = browp[k + 1];
        // 8 args: (neg_a, A, neg_b, B, c_mod, C, reuse_a, reuse_b)
        acc = __builtin_amdgcn_wmma_f32_16x16x4_f32(false, a, false, b,
                                                    (short)0, acc, false, false);
    }

    // D layout: VGPR v, lanes 0-15: M=v, N=lane ; lanes 16-31: M=v+8, N=lane-16
    const int o = lane & 15;
    if (o < 2 * P) {
        const int   comp  = o & 1;              // 0 = x-dir (W), 1 = y-dir (H)
        const int   p     = o >> 1;
        const float bias  = bo[o];
        const float dim   = (comp == 0) ? (float)Wimg : (float)Himg;
        const float inorm = 1.0f / dim;
        const int   mbase = (lane >> 4) << 3;   // 0 or 8
        #pragma unroll
        for (int v = 0; v < 8; ++v) {
            const int r = r0 + v + mbase;
            if (r >= R) continue;
            const int n = r % N;
            const int h = n / Wimg, w = n % Wimg;
            const float ref = (comp == 0) ? ((float)w + 0.5f) * (1.0f / (float)Wimg)
                                          : ((float)h + 0.5f) * (1.0f / (float)Himg);
            const float loc = ref + (acc[v] + bias) * inorm;
            samploc[((size_t)r * P + p) * 2 + comp] = loc;
        }
    }
}

// ---------------------------------------------------------------------------
// Kernel 2: bilinear zero-padded sampling + combine. One wave per (b,n);
// lane t owns channels 4t..4t+3 (float4) so each corner gather is a fully
// coalesced wave-wide 512B contiguous burst from x / deltaA. Invalid corners
// use clamped in-bounds addresses with weight forced to zero (branch-free).
// out[r,c] = (1/P) * sum_p feat_p[c] * (wt_p[c] - deltaA[r,c])
// ---------------------------------------------------------------------------
__global__ __launch_bounds__(256)
void sao_sample(const float* __restrict__ X,
                const float* __restrict__ dA,
                const float* __restrict__ samploc,
                const int* __restrict__ Hp,
                const int* __restrict__ Wp,
                float* __restrict__ out,
                int R, int C, int P)
{
    const int lane = threadIdx.x & 31;
    const int wid  = threadIdx.x >> 5;
    const int r    = blockIdx.x * 8 + wid;      // r = b*N + n
    if (r >= R) return;

    const int Himg = *Hp, Wimg = *Wp;
    const int N = Himg * Wimg;
    const int b = r / N;
    const size_t rowbase = (size_t)b * N;       // first row of this image
    const float fW = (float)Wimg, fH = (float)Himg;

    for (int c4 = lane * 4; c4 < C; c4 += 128) {
        const float4 da = *(const float4*)(dA + (size_t)r * C + c4);
        float accx = 0.f, accy = 0.f, accz = 0.f, accw = 0.f;

        for (int p = 0; p < P; ++p) {
            const float locx = samploc[((size_t)r * P + p) * 2 + 0];
            const float locy = samploc[((size_t)r * P + p) * 2 + 1];
            // x = ((2*loc - 1 + 1)*W - 1)*0.5 = loc*W - 0.5
            const float xf = locx * fW - 0.5f;
            const float yf = locy * fH - 0.5f;
            const float x0f = floorf(xf), y0f = floorf(yf);
            const float wx1 = xf - x0f, wx0 = 1.0f - wx1;
            const float wy1 = yf - y0f, wy0 = 1.0f - wy1;
            const int x0 = (int)x0f, y0 = (int)y0f;

            float fx0 = 0.f, fx1 = 0.f, fx2 = 0.f, fx3 = 0.f;   // feat accum
            float wt0 = 0.f, wt1 = 0.f, wt2 = 0.f, wt3 = 0.f;   // wt accum
            #pragma unroll
            for (int dy = 0; dy < 2; ++dy) {
                const int iy = y0 + dy;
                const float wy = dy ? wy1 : wy0;
                const bool vy = (iy >= 0) && (iy < Himg);
                const int iyc = min(max(iy, 0), Himg - 1);
                #pragma unroll
                for (int dx = 0; dx < 2; ++dx) {
                    const int ix = x0 + dx;
                    const bool v = vy && (ix >= 0) && (ix < Wimg);
                    const int ixc = min(max(ix, 0), Wimg - 1);
                    const float wgt = (v ? 1.0f : 0.0f) * wy * (dx ? wx1 : wx0);
                    const size_t row = rowbase + (size_t)iyc * Wimg + ixc;
                    const float4 fv = *(const float4*)(X  + row * C + c4);
                    const float4 dv = *(const float4*)(dA + row * C + c4);
                    fx0 += wgt * fv.x; fx1 += wgt * fv.y;
                    fx2 += wgt * fv.z; fx3 += wgt * fv.w;
                    wt0 += wgt * dv.x; wt1 += wgt * dv.y;
                    wt2 += wgt * dv.z; wt3 += wgt * dv.w;
                }
            }
            accx += fx0 * (wt0 - da.x);
            accy += fx1 * (wt1 - da.y);
            accz += fx2 * (wt2 - da.z);
            accw += fx3 * (wt3 - da.w);
        }

        const float s = 1.0f / (float)P;
        float4 o = make_float4(accx * s, accy * s, accz * s, accw * s);
        *(float4*)(out + (size_t)r * C + c4) = o;
    }
}

extern "C" void kernel_launch(void* const* d_in, const int* in_sizes, int n_in,
                              void* d_out, int out_size, void* d_ws, size_t ws_size,
                              hipStream_t stream) {
    // Inputs: x, deltaA, H, W, Wo, bo
    const float* x   = (const float*)d_in[0];
    const float* dA  = (const float*)d_in[1];
    const int*   Hp  = (const int*)d_in[2];
    const int*   Wp  = (const int*)d_in[3];
    const float* Wo  = (const float*)d_in[4];
    const float* bo  = (const float*)d_in[5];

    // Derive shapes from element counts (H,W values stay on device):
    const int twoP = in_sizes[5];               // 2P
    const int P    = twoP / 2;
    const int C    = in_sizes[4] / twoP;        // Wo is (2P, C)
    const int R    = in_sizes[0] / C;           // B*N flattened rows

    float* out     = (float*)d_out;             // R*C floats
    float* samploc = (float*)d_out + (size_t)R * C;  // R*P*2 floats

    // Kernel 1: WMMA offset GEMM + sampling-location write (8 waves/block)
    const int ntiles = (R + 15) / 16;
    const int blk1   = (ntiles + 7) / 8;
    sao_off_wmma<<<blk1, 256, 0, stream>>>(x, Wo, bo, Hp, Wp, samploc, R, C, P, ntiles);

    // Kernel 2: bilinear gather + combine (8 waves/block, 1 wave per (b,n))
    const int blk2 = (R + 7) / 8;
    sao_sample<<<blk2, 256, 0, stream>>>(x, dA, samploc, Hp, Wp, out, R, C, P);
}